// GCN_26242250178564
// MI455X (gfx1250) — compile-verified
//
#include <hip/hip_runtime.h>

#define D_FEAT 256
#define N_NODES 50000
#define N_EDGES 800000

typedef float v2f __attribute__((ext_vector_type(2)));
typedef float v8f __attribute__((ext_vector_type(8)));
typedef int   v4i __attribute__((ext_vector_type(4)));

#define AS1 __attribute__((address_space(1)))
#define AS3 __attribute__((address_space(3)))

#if __has_builtin(__builtin_amdgcn_global_load_async_to_lds_b128) && \
    __has_builtin(__builtin_amdgcn_global_load_async_to_lds_b32)
#define HAVE_ASYNC_LDS 1
#endif

template <int N>
__device__ __forceinline__ void wait_asynccnt() {
#if __has_builtin(__builtin_amdgcn_s_wait_asynccnt)
    __builtin_amdgcn_s_wait_asynccnt(N);
#else
    asm volatile("s_wait_asynccnt %0" ::"i"(N) : "memory");
#endif
}

// ---------------- init kernels ----------------

__global__ void init_deg_kernel(float* __restrict__ deg, int n) {
    int i = blockIdx.x * blockDim.x + threadIdx.x;
    if (i < n) deg[i] = 1.0f;  // self-loop contributes 1 to every node's degree
}

__global__ void zero_out_kernel(float4* __restrict__ p, long n4) {
    long i = (long)blockIdx.x * blockDim.x + threadIdx.x;
    if (i < n4) p[i] = make_float4(0.f, 0.f, 0.f, 0.f);
}

__global__ void deg_accum_kernel(const int* __restrict__ ei, float* __restrict__ deg, int E) {
    int e = blockIdx.x * blockDim.x + threadIdx.x;
    if (e < E) {
        int dst = ei[2 * e + 1];
        unsafeAtomicAdd(&deg[dst], 1.0f);
    }
}

__global__ void dinv_kernel(float* __restrict__ deg, int n) {
    int i = blockIdx.x * blockDim.x + threadIdx.x;
    if (i < n) deg[i] = rsqrtf(deg[i]);  // deg >= 1 always (self-loop)
}

// ---------------- GEMM: H = X @ W  (M=50000, N=256, K=256), f32 WMMA ----------------
// Block = 256 threads = 8 waves; block computes 16 rows x 256 cols.
// Wave w owns cols [32w, 32w+32): two 16x16 f32 accumulators.
// K loop: 16 chunks of K=16, double-buffered in LDS via async global->LDS copies
// (GLOBAL_LOAD_ASYNC_TO_LDS_B128/B32, ASYNCcnt) with a synchronous fallback.

#define KCHUNK 16
#define NCHUNK (D_FEAT / KCHUNK)

__launch_bounds__(256)
__global__ void gemm_wmma_kernel(const float* __restrict__ X,
                                 const float* __restrict__ W,
                                 float* __restrict__ H) {
    __shared__ float As[2][16][16];       // [2][16 rows][16 k]
    __shared__ float Bs[2][16][256];      // [2][16 k][256 cols]

    const int t    = threadIdx.x;
    const int wave = t >> 5;
    const int lane = t & 31;
    const int half = lane >> 4;   // 0: lanes 0-15, 1: lanes 16-31
    const int r    = lane & 15;
    const int row0 = blockIdx.x * 16;     // 50000/16 = 3125 exact, EXEC all-ones
    const int n0   = wave * 32;

    // Per-thread staging coordinates (uniform across waves -> uniform ASYNCcnt)
    const int a_row  = t >> 4;            // 0..15
    const int a_col  = t & 15;            // 0..15
    const int b_row0 = t >> 6;            // float4 linear: f = t + 256*i
    const int b_c4   = t & 63;

    v8f c0 = {};
    v8f c1 = {};

#ifdef HAVE_ASYNC_LDS
    // ---- async double-buffered pipeline ----
    // issue chunk 'c' into buffer c&1: 4 x b128 (B) + 1 x b32 (A) per thread
    auto issue = [&](int c) {
        const int k0  = c * KCHUNK;
        const int buf = c & 1;
        #pragma unroll
        for (int i = 0; i < 4; ++i) {
            int j = b_row0 + 4 * i;                  // k-row within chunk (f>>6)
            const float* g = W + (size_t)(k0 + j) * D_FEAT + b_c4 * 4;
            float*       l = &Bs[buf][j][b_c4 * 4];
            __builtin_amdgcn_global_load_async_to_lds_b128(
                (AS1 v4i*)g, (AS3 v4i*)l, 0, 0);
        }
        {
            const float* g = X + (size_t)(row0 + a_row) * D_FEAT + k0 + a_col;
            float*       l = &As[buf][a_row][a_col];
            __builtin_amdgcn_global_load_async_to_lds_b32(
                (AS1 int*)g, (AS3 int*)l, 0, 0);
        }
    };

    issue(0);
    for (int c = 0; c < NCHUNK; ++c) {
        const int buf = c & 1;
        if (c + 1 < NCHUNK) {
            issue(c + 1);          // overwrites buffer retired at end of iter c-1
            wait_asynccnt<5>();    // chunk c complete; chunk c+1 (5 ops) in flight
        } else {
            wait_asynccnt<0>();
        }
        __syncthreads();

        #pragma unroll
        for (int kk = 0; kk < KCHUNK; kk += 4) {
            v2f a;
            a.x = As[buf][r][kk + 2 * half + 0];
            a.y = As[buf][r][kk + 2 * half + 1];
            v2f b0;
            b0.x = Bs[buf][kk + 2 * half + 0][n0 + r];
            b0.y = Bs[buf][kk + 2 * half + 1][n0 + r];
            v2f b1;
            b1.x = Bs[buf][kk + 2 * half + 0][n0 + 16 + r];
            b1.y = Bs[buf][kk + 2 * half + 1][n0 + 16 + r];
            c0 = __builtin_amdgcn_wmma_f32_16x16x4_f32(false, a, false, b0,
                                                       (short)0, c0, false, false);
            c1 = __builtin_amdgcn_wmma_f32_16x16x4_f32(false, a, false, b1,
                                                       (short)0, c1, false, false);
        }
        __syncthreads();           // retire buf before it is overwritten at c+2
    }
#else
    // ---- synchronous fallback ----
    for (int c = 0; c < NCHUNK; ++c) {
        const int k0  = c * KCHUNK;
        const int buf = c & 1;
        #pragma unroll
        for (int i = 0; i < 4; ++i) {
            int f  = t + 256 * i;
            int j  = f >> 6;
            int c4 = f & 63;
            *(float4*)&Bs[buf][j][c4 * 4] =
                *(const float4*)(W + (size_t)(k0 + j) * D_FEAT + c4 * 4);
        }
        As[buf][a_row][a_col] = X[(size_t)(row0 + a_row) * D_FEAT + k0 + a_col];
        __syncthreads();

        #pragma unroll
        for (int kk = 0; kk < KCHUNK; kk += 4) {
            v2f a;
            a.x = As[buf][r][kk + 2 * half + 0];
            a.y = As[buf][r][kk + 2 * half + 1];
            v2f b0;
            b0.x = Bs[buf][kk + 2 * half + 0][n0 + r];
            b0.y = Bs[buf][kk + 2 * half + 1][n0 + r];
            v2f b1;
            b1.x = Bs[buf][kk + 2 * half + 0][n0 + 16 + r];
            b1.y = Bs[buf][kk + 2 * half + 1][n0 + 16 + r];
            c0 = __builtin_amdgcn_wmma_f32_16x16x4_f32(false, a, false, b0,
                                                       (short)0, c0, false, false);
            c1 = __builtin_amdgcn_wmma_f32_16x16x4_f32(false, a, false, b1,
                                                       (short)0, c1, false, false);
        }
        __syncthreads();
    }
#endif

    // C/D layout: VGPR v, lanes 0-15 -> row v, lanes 16-31 -> row v+8; col = lane&15
    #pragma unroll
    for (int v = 0; v < 8; ++v) {
        int m = row0 + v + 8 * half;
        H[(size_t)m * D_FEAT + n0 + r]      = c0[v];
        H[(size_t)m * D_FEAT + n0 + 16 + r] = c1[v];
    }
}

// ---------------- edge scatter: out[dst] += dinv[src]*dinv[dst] * h[src] ----------------

__launch_bounds__(256)
__global__ void scatter_kernel(const float* __restrict__ H,
                               const int* __restrict__ ei,
                               const float* __restrict__ dinv,
                               float* __restrict__ out, int E) {
    int gw   = (blockIdx.x * blockDim.x + threadIdx.x) >> 5;  // global wave id = edge id
    int lane = threadIdx.x & 31;
    if (gw >= E) return;
    int src = ei[2 * gw + 0];
    int dst = ei[2 * gw + 1];
    float norm = dinv[src] * dinv[dst];
    const float* hs = H + (size_t)src * D_FEAT;
    float* od       = out + (size_t)dst * D_FEAT;
    #pragma unroll
    for (int i = 0; i < 8; ++i) {
        int d = lane + 32 * i;
        unsafeAtomicAdd(&od[d], norm * hs[d]);
    }
}

// ---------------- finalize: out = relu(out + dinv^2 * h + b) ----------------

__launch_bounds__(256)
__global__ void finalize_kernel(float* __restrict__ out,
                                const float* __restrict__ H,
                                const float* __restrict__ dinv,
                                const float* __restrict__ b) {
    long i   = (long)blockIdx.x * blockDim.x + threadIdx.x;  // over N*D (exact)
    int node = (int)(i >> 8);
    int d    = (int)(i & 255);
    float s  = dinv[node];
    float v  = out[i] + s * s * H[i] + b[d];
    out[i]   = fmaxf(v, 0.0f);
}

// ---------------- launch ----------------

extern "C" void kernel_launch(void* const* d_in, const int* in_sizes, int n_in,
                              void* d_out, int out_size, void* d_ws, size_t ws_size,
                              hipStream_t stream) {
    const float* obj_vecs = (const float*)d_in[0];
    const float* rel_vecs = (const float*)d_in[1];
    const int*   ei       = (const int*)d_in[2];   // [E,2] int32
    const float* W_obj    = (const float*)d_in[3];
    const float* b_obj    = (const float*)d_in[4];
    const float* W_rel    = (const float*)d_in[5];
    const float* b_rel    = (const float*)d_in[6];

    float* out0 = (float*)d_out;                                // obj_out [N,D]
    float* out1 = out0 + (size_t)N_NODES * D_FEAT;              // rel_out [N,D]

    float* h   = (float*)d_ws;                                  // [N,D] scratch
    float* dnv = h + (size_t)N_NODES * D_FEAT;                  // [N] deg -> dinv in place

    const int  nd = N_NODES * D_FEAT;            // 12.8M, exact multiple of 256
    const long n4 = (long)2 * nd / 4;            // float4 count for both outputs

    init_deg_kernel<<<(N_NODES + 255) / 256, 256, 0, stream>>>(dnv, N_NODES);
    zero_out_kernel<<<(int)((n4 + 255) / 256), 256, 0, stream>>>((float4*)d_out, n4);
    deg_accum_kernel<<<(N_EDGES + 255) / 256, 256, 0, stream>>>(ei, dnv, N_EDGES);
    dinv_kernel<<<(N_NODES + 255) / 256, 256, 0, stream>>>(dnv, N_NODES);

    const int gemm_blocks    = N_NODES / 16;            // 3125, exact
    const int scatter_blocks = (N_EDGES + 7) / 8;       // 8 waves/block, 1 edge/wave
    const int fin_blocks     = nd / 256;                // exact

    // conv 1: obj
    gemm_wmma_kernel<<<gemm_blocks, 256, 0, stream>>>(obj_vecs, W_obj, h);
    scatter_kernel<<<scatter_blocks, 256, 0, stream>>>(h, ei, dnv, out0, N_EDGES);
    finalize_kernel<<<fin_blocks, 256, 0, stream>>>(out0, h, dnv, b_obj);

    // conv 2: rel (reuses h; stream order serializes)
    gemm_wmma_kernel<<<gemm_blocks, 256, 0, stream>>>(rel_vecs, W_rel, h);
    scatter_kernel<<<scatter_blocks, 256, 0, stream>>>(h, ei, dnv, out1, N_EDGES);
    finalize_kernel<<<fin_blocks, 256, 0, stream>>>(out1, h, dnv, b_rel);
}